// BasicGCNBlock_37769942401557
// MI455X (gfx1250) — compile-verified
//
#include <hip/hip_runtime.h>

typedef float v2f __attribute__((ext_vector_type(2)));
typedef float v8f __attribute__((ext_vector_type(8)));

#define N_NODES 100000
#define N_EDGES 1600000
#define D 64
#define LN_EPS 1e-5f

// ---------------------------------------------------------------------------
// Zero the accumulator (d_out) — needed every launch for determinism.
__global__ __launch_bounds__(256) void zero_acc_kernel(float* __restrict__ acc) {
    size_t i = (size_t)blockIdx.x * blockDim.x + threadIdx.x;
    if (i < (size_t)N_NODES * D) acc[i] = 0.0f;
}

// Degree init: self-loop contributes 1 to every node's in-degree.
__global__ __launch_bounds__(256) void init_deg_kernel(float* __restrict__ deg) {
    int i = blockIdx.x * blockDim.x + threadIdx.x;
    if (i < N_NODES) deg[i] = 1.0f;
}

// deg[dst] += 1 for every real edge.
__global__ __launch_bounds__(256) void degree_kernel(const long long* __restrict__ ei,
                                                     float* __restrict__ deg) {
    int e = blockIdx.x * blockDim.x + threadIdx.x;
    if (e < N_EDGES) {
        int d = (int)ei[(size_t)N_EDGES + e];
        unsafeAtomicAdd(&deg[d], 1.0f);
    }
}

// deg -> dinv = 1/sqrt(deg) in place (deg >= 1 always).
__global__ __launch_bounds__(256) void rsqrt_kernel(float* __restrict__ deg) {
    int i = blockIdx.x * blockDim.x + threadIdx.x;
    if (i < N_NODES) deg[i] = rsqrtf(deg[i]);
}

// ---------------------------------------------------------------------------
// h = x @ W via V_WMMA_F32_16X16X4_F32.
// One wave computes a 16(rows)x16(cols) tile; 4 waves/block -> 64 cols;
// grid.x = N/16 = 6250 row blocks. K=64 -> 16 chained WMMAs.
__global__ __launch_bounds__(128) void gemm_wmma_kernel(const float* __restrict__ x,
                                                        const float* __restrict__ W,
                                                        float* __restrict__ h) {
    const int wave = threadIdx.x >> 5;
    const int lane = threadIdx.x & 31;
    const int half = lane >> 4;      // 0: lanes 0-15, 1: lanes 16-31
    const int l    = lane & 15;
    const int row0 = blockIdx.x << 4;
    const int col0 = wave << 4;

    // A fragment source: row (row0+l), K base = half*2 (float2-aligned)
    const float2* arow = (const float2*)(x + (size_t)(row0 + l) * D + half * 2);
    // B fragment source: W[K][col0+l], K base = half*2
    const float*  bcol = W + (size_t)(half * 2) * D + col0 + l;

    v8f acc = {};
#pragma unroll
    for (int kk = 0; kk < 16; ++kk) {
        float2 af = arow[2 * kk];                 // x[row][4kk+2h], x[row][4kk+2h+1]
        v2f a = {af.x, af.y};
        v2f b = {bcol[(size_t)(4 * kk) * D],      // W[4kk+2h  ][col]
                 bcol[(size_t)(4 * kk + 1) * D]}; // W[4kk+2h+1][col]
        acc = __builtin_amdgcn_wmma_f32_16x16x4_f32(
            /*neg_a=*/false, a, /*neg_b=*/false, b,
            /*c_mod=*/(short)0, acc, /*reuse_a=*/false, /*reuse_b=*/false);
    }

    // D layout: VGPR r -> M = r + 8*half, N = l
    float* hout = h + (size_t)(row0 + 8 * half) * D + col0 + l;
#pragma unroll
    for (int r = 0; r < 8; ++r)
        hout[(size_t)r * D] = acc[r];
}

// ---------------------------------------------------------------------------
// Edge scatter: acc[dst] += h[src] * dinv[src]*dinv[dst].
// 16 threads per edge, 4 consecutive floats (float4) per thread.
// h and acc are L2-resident (25.6 MB each, 192 MB L2).
__global__ __launch_bounds__(256) void edge_scatter_kernel(const long long* __restrict__ ei,
                                                           const float* __restrict__ h,
                                                           const float* __restrict__ dinv,
                                                           float* __restrict__ acc) {
    long long t = (long long)blockIdx.x * blockDim.x + threadIdx.x;
    long long e = t >> 4;
    if (e >= N_EDGES) return;
    int fi = ((int)t & 15) * 4;

    long long s = ei[e];
    long long d = ei[(size_t)N_EDGES + e];
    float w = dinv[s] * dinv[d];

    float4 hv = *(const float4*)(h + (size_t)s * D + fi);
    float* o = acc + (size_t)d * D + fi;
    unsafeAtomicAdd(o + 0, hv.x * w);
    unsafeAtomicAdd(o + 1, hv.y * w);
    unsafeAtomicAdd(o + 2, hv.z * w);
    unsafeAtomicAdd(o + 3, hv.w * w);
}

// ---------------------------------------------------------------------------
// Finalize (in place on acc == d_out):
//   v = acc + h*dinv^2 (self loop) + b ; ReLU ; LayerNorm(gamma, beta)
// One wave32 per node, 2 features per lane, shuffle reduction for mean/var.
__global__ __launch_bounds__(256) void finalize_kernel(float* __restrict__ acc,
                                                       const float* __restrict__ h,
                                                       const float* __restrict__ dinv,
                                                       const float* __restrict__ bias,
                                                       const float* __restrict__ gamma,
                                                       const float* __restrict__ beta) {
    int wave = threadIdx.x >> 5;
    int lane = threadIdx.x & 31;
    int node = blockIdx.x * 8 + wave;
    if (node >= N_NODES) return;

    float di = dinv[node];
    float selfw = di * di;
    int f = lane * 2;

    float2 av = *(const float2*)(acc + (size_t)node * D + f);
    float2 hv = *(const float2*)(h + (size_t)node * D + f);
    float v0 = fmaxf(av.x + hv.x * selfw + bias[f], 0.0f);
    float v1 = fmaxf(av.y + hv.y * selfw + bias[f + 1], 0.0f);

    float s  = v0 + v1;
    float sq = v0 * v0 + v1 * v1;
#pragma unroll
    for (int m = 16; m >= 1; m >>= 1) {
        s  += __shfl_xor(s, m, 32);
        sq += __shfl_xor(sq, m, 32);
    }
    float mean = s * (1.0f / D);
    float var  = sq * (1.0f / D) - mean * mean;
    float inv  = rsqrtf(var + LN_EPS);

    float2 o;
    o.x = (v0 - mean) * inv * gamma[f] + beta[f];
    o.y = (v1 - mean) * inv * gamma[f + 1] + beta[f + 1];
    *(float2*)(acc + (size_t)node * D + f) = o;
}

// ---------------------------------------------------------------------------
extern "C" void kernel_launch(void* const* d_in, const int* in_sizes, int n_in,
                              void* d_out, int out_size, void* d_ws, size_t ws_size,
                              hipStream_t stream) {
    const float*     x     = (const float*)d_in[0];
    const long long* ei    = (const long long*)d_in[1];   // int64 [2, E]
    const float*     W     = (const float*)d_in[2];
    const float*     bias  = (const float*)d_in[3];
    const float*     gamma = (const float*)d_in[4];
    const float*     beta  = (const float*)d_in[5];
    float*           out   = (float*)d_out;               // [N, 64], also the accumulator

    // Workspace layout: dinv (N floats) | h (N*64 floats)  -> ~26 MB
    float* dinv = (float*)d_ws;
    float* h    = dinv + N_NODES;   // 400000 B offset, 16B aligned

    // 1) zero accumulator (d_out)
    zero_acc_kernel<<<(N_NODES * D + 255) / 256, 256, 0, stream>>>(out);
    // 2) degree = 1 (self loop) + scatter count, then -> rsqrt
    init_deg_kernel<<<(N_NODES + 255) / 256, 256, 0, stream>>>(dinv);
    degree_kernel<<<(N_EDGES + 255) / 256, 256, 0, stream>>>(ei, dinv);
    rsqrt_kernel<<<(N_NODES + 255) / 256, 256, 0, stream>>>(dinv);
    // 3) h = x @ W (WMMA f32)
    gemm_wmma_kernel<<<N_NODES / 16, 128, 0, stream>>>(x, W, h);
    // 4) edge scatter-add into d_out
    edge_scatter_kernel<<<(N_EDGES * 16 + 255) / 256, 256, 0, stream>>>(ei, h, dinv, out);
    // 5) self loop + bias + ReLU + LayerNorm, in place
    finalize_kernel<<<(N_NODES + 7) / 8, 256, 0, stream>>>(out, h, dinv, bias, gamma, beta);
}